// GAEncoder_36627481101159
// MI455X (gfx1250) — compile-verified
//
#include <hip/hip_runtime.h>

typedef __attribute__((ext_vector_type(2))) float v2f;
typedef __attribute__((ext_vector_type(8))) float v8f;

#define GCN_N 100000

// ---------------- degree / normalization ----------------

__global__ void deg_kernel(const long long* __restrict__ dst, float* __restrict__ deg, int nE) {
    int e = blockIdx.x * blockDim.x + threadIdx.x;
    if (e < nE) {
        atomicAdd(&deg[(int)dst[e]], 1.0f);
    }
}

__global__ void dinv_kernel(float* __restrict__ deg, int n) {
    int i = blockIdx.x * blockDim.x + threadIdx.x;
    if (i < n) {
        // +1 self-loop => deg >= 1 always, no zero guard needed
        deg[i] = rsqrtf(deg[i] + 1.0f);
    }
}

// ---------------- fp32 WMMA GEMM: H[N x COUT] = X[N x K] @ W[K x COUT] ----------------
// One wave computes one 16x16 output tile with K/4 V_WMMA_F32_16X16X4_F32 ops.

template<int K, int COUT>
__global__ __launch_bounds__(256) void gemm_wmma(const float* __restrict__ X,
                                                 const float* __restrict__ W,
                                                 float* __restrict__ H,
                                                 int nRowTiles) {
    const int lane = threadIdx.x & 31;
    const int wave = (blockIdx.x * blockDim.x + threadIdx.x) >> 5;
    const int nColTiles = COUT / 16;
    const int nTiles = nRowTiles * nColTiles;
    if (wave >= nTiles) return;                 // wave-uniform: EXEC all-1s below
    const int mTile = wave / nColTiles;
    const int nTile = wave - mTile * nColTiles;
    const int m0 = mTile * 16;
    const int n0 = nTile * 16;
    const int half = lane >> 4;                 // 0: lanes 0-15, 1: lanes 16-31
    const int l15 = lane & 15;

    v8f c = {};
#pragma unroll
    for (int k0 = 0; k0 < K; k0 += 4) {
        const int kb = k0 + half * 2;
        // A fragment (16x4): lane half selects K pair {0,1} vs {2,3}
        const float2 av = *(const float2*)(X + (size_t)(m0 + l15) * K + kb);
        v2f a; a.x = av.x; a.y = av.y;
        // B fragment (4x16): row kb / kb+1, col n0+l15
        v2f b;
        b.x = W[(size_t)kb * COUT + n0 + l15];
        b.y = W[(size_t)(kb + 1) * COUT + n0 + l15];
        c = __builtin_amdgcn_wmma_f32_16x16x4_f32(false, a, false, b,
                                                  (short)0, c, false, false);
    }
#pragma unroll
    for (int v = 0; v < 8; ++v) {
        const int row = m0 + v + half * 8;      // C/D layout: VGPR v -> M=v (+8 upper half)
        H[(size_t)row * COUT + n0 + l15] = c[v];
    }
}

// ---------------- edge scatter: agg[dst] += h[src] * dinv[src]*dinv[dst] ----------------
// One wave per edge (grid-stride); lane-parallel over features.

template<int C>
__global__ __launch_bounds__(256) void scatter_kernel(const float* __restrict__ h,
                                                      const long long* __restrict__ src,
                                                      const long long* __restrict__ dst,
                                                      const float* __restrict__ dinv,
                                                      float* __restrict__ agg, int nE) {
    const int lane = threadIdx.x & 31;
    const int wave = (blockIdx.x * blockDim.x + threadIdx.x) >> 5;
    const int nWaves = (gridDim.x * blockDim.x) >> 5;
    for (int e = wave; e < nE; e += nWaves) {
        const int s = (int)src[e];
        const int d = (int)dst[e];
        const float norm = dinv[s] * dinv[d];
#pragma unroll
        for (int f = lane; f < C; f += 32) {
            atomicAdd(&agg[(size_t)d * C + f], h[(size_t)s * C + f] * norm);
        }
    }
}

// ---------------- finalize: agg += dinv^2 * h (self-loop) + bias [, ReLU] ----------------

template<bool RELU>
__global__ void finalize_kernel(float* __restrict__ agg, const float* __restrict__ h,
                                const float* __restrict__ dinv, const float* __restrict__ bias,
                                int total, int C) {
    int i = blockIdx.x * blockDim.x + threadIdx.x;
    if (i < total) {
        int node = i / C;
        int f = i - node * C;
        float di = dinv[node];
        float v = agg[i] + di * di * h[i] + bias[f];
        if (RELU) v = fmaxf(v, 0.0f);
        agg[i] = v;
    }
}

extern "C" void kernel_launch(void* const* d_in, const int* in_sizes, int n_in,
                              void* d_out, int out_size, void* d_ws, size_t ws_size,
                              hipStream_t stream) {
    const float*     x   = (const float*)d_in[0];
    const long long* ei  = (const long long*)d_in[1];   // int64 [2, E]
    const float*     W1  = (const float*)d_in[2];
    const float*     b1  = (const float*)d_in[3];
    const float*     W2  = (const float*)d_in[4];
    const float*     b2  = (const float*)d_in[5];
    float* out = (float*)d_out;

    const int nE = in_sizes[1] / 2;
    const long long* src = ei;
    const long long* dst = ei + nE;

    const int N = GCN_N;
    const int IN_C = 64, HID = 64, OUT_C = 32;
    (void)IN_C;

    float* ws   = (float*)d_ws;
    float* dinv = ws;                                   // N floats (padded region)
    float* h1   = ws + 102400;                          // N*64
    float* agg1 = h1 + (size_t)N * HID;                 // N*64 (reused as ReLU'd h1)
    float* h2   = agg1 + (size_t)N * HID;               // N*32

    hipMemsetAsync(dinv, 0, (size_t)N * sizeof(float), stream);
    hipMemsetAsync(agg1, 0, (size_t)N * HID * sizeof(float), stream);
    hipMemsetAsync(out,  0, (size_t)N * OUT_C * sizeof(float), stream);

    deg_kernel<<<(nE + 255) / 256, 256, 0, stream>>>(dst, dinv, nE);
    dinv_kernel<<<(N + 255) / 256, 256, 0, stream>>>(dinv, N);

    // Layer 1: h1 = x @ W1 ; aggregate ; +b1 ; ReLU (in place into agg1)
    {
        const int nRowTiles = N / 16;                   // 6250 (exact)
        const int nTiles = nRowTiles * (HID / 16);      // 25000 waves
        const int threads = nTiles * 32;
        gemm_wmma<64, 64><<<(threads + 255) / 256, 256, 0, stream>>>(x, W1, h1, nRowTiles);
    }
    scatter_kernel<64><<<4096, 256, 0, stream>>>(h1, src, dst, dinv, agg1, nE);
    finalize_kernel<true><<<((N * HID) + 255) / 256, 256, 0, stream>>>(
        agg1, h1, dinv, b1, N * HID, HID);

    // Layer 2: h2 = relu_out @ W2 ; aggregate into d_out ; +b2
    {
        const int nRowTiles = N / 16;
        const int nTiles = nRowTiles * (OUT_C / 16);    // 12500 waves
        const int threads = nTiles * 32;
        gemm_wmma<64, 32><<<(threads + 255) / 256, 256, 0, stream>>>(agg1, W2, h2, nRowTiles);
    }
    scatter_kernel<32><<<4096, 256, 0, stream>>>(h2, src, dst, dinv, out, nE);
    finalize_kernel<false><<<((N * OUT_C) + 255) / 256, 256, 0, stream>>>(
        out, h2, dinv, b2, N * OUT_C, OUT_C);
}